// SpGraphAttentionLayer_48103633715623
// MI455X (gfx1250) — compile-verified
//
#include <hip/hip_runtime.h>

#define F_IN   512
#define F_OUT  256
#define ALPHA  0.2f
#define GAT_EPS 9e-15f
#define LDS_PITCH 520   // 512 + 8 pad: stride mod 64 banks = 8 -> conflict-free column reads

typedef __attribute__((ext_vector_type(2))) float v2f;
typedef __attribute__((ext_vector_type(8))) float v8f;

// ---------------------------------------------------------------------------
// Kernel 1: h = x @ W + bias via V_WMMA_F32_16X16X4_F32.
// One block = 8 waves handles 16 rows of x (staged in LDS) x all 256 cols.
// Each wave produces two 16x16 tiles, K-loop in steps of 4.
// ---------------------------------------------------------------------------
__global__ __launch_bounds__(256) void gat_gemm(const float* __restrict__ x,
                                                const float* __restrict__ W,
                                                const float* __restrict__ bias,
                                                float* __restrict__ h) {
    __shared__ float xs[16 * LDS_PITCH];

    const int tid = threadIdx.x;
    const int rowBlock = blockIdx.x;                 // 16 rows per block

    // Cooperative stage of the 16x512 f32 slab (2048 float4, 8 per thread).
    const float4* xv = (const float4*)(x + (size_t)rowBlock * 16 * F_IN);
    #pragma unroll
    for (int q = tid; q < 2048; q += 256) {
        const int r  = q >> 7;                       // row 0..15
        const int cc = (q & 127) << 2;               // col 0..508
        float4 v = xv[q];                            // rows are contiguous in x
        *(float4*)&xs[r * LDS_PITCH + cc] = v;
    }
    __syncthreads();

    const int lane = tid & 31;
    const int wave = tid >> 5;
    const int half = lane >> 4;                      // which 16-lane half
    const int l16  = lane & 15;
    const int koff = half * 2;                       // K sub-offset per ISA A/B layout

    const float2* xrow = (const float2*)&xs[l16 * LDS_PITCH];

    #pragma unroll
    for (int t = 0; t < 2; ++t) {
        const int ct  = wave * 2 + t;                // column tile 0..15
        const int col = ct * 16 + l16;
        const float* Wc = W + col;

        v8f c = {0.f, 0.f, 0.f, 0.f, 0.f, 0.f, 0.f, 0.f};

        #pragma unroll 4
        for (int k = 0; k < F_IN; k += 4) {
            // A: 16x4 slice of x.  lanes 0-15 -> K=k,k+1 ; lanes 16-31 -> K=k+2,k+3
            float2 af = xrow[(k + koff) >> 1];
            v2f a; a.x = af.x; a.y = af.y;
            // B: 4x16 slice of W.  lane holds column N=l16, VGPRs walk K.
            v2f b;
            b.x = Wc[(size_t)(k + koff)     * F_OUT];
            b.y = Wc[(size_t)(k + koff + 1) * F_OUT];
            __builtin_prefetch(&Wc[(size_t)(k + 16) * F_OUT], 0, 0);
            c = __builtin_amdgcn_wmma_f32_16x16x4_f32(
                    /*neg_a=*/false, a, /*neg_b=*/false, b,
                    /*c_mod=*/(short)0, c, /*reuse_a=*/false, /*reuse_b=*/false);
        }

        // D layout: VGPR r holds (row = r + 8*half, col = l16).
        float* hb = h + (size_t)rowBlock * 16 * F_OUT;
        const float bv = bias[col];
        #pragma unroll
        for (int r = 0; r < 8; ++r) {
            const int row = r + half * 8;
            hb[(size_t)row * F_OUT + col] = c[r] + bv;
        }
    }
}

// ---------------------------------------------------------------------------
// Kernel 2: per-node attention coefficients  e_src[n]=h[n].a_src, e_dst[n]=h[n].a_dst
// One wave per node, lane-strided over 256 features, shuffle reduction.
// ---------------------------------------------------------------------------
__global__ __launch_bounds__(256) void gat_coef(const float* __restrict__ h,
                                                const float* __restrict__ a,
                                                float* __restrict__ esrc,
                                                float* __restrict__ edst, int N) {
    const int node = blockIdx.x * 8 + (threadIdx.x >> 5);
    const int lane = threadIdx.x & 31;
    if (node >= N) return;
    const float* hr = h + (size_t)node * F_OUT;
    float ss = 0.f, sd = 0.f;
    #pragma unroll
    for (int t = 0; t < 8; ++t) {
        const int f = lane + 32 * t;
        const float hv = hr[f];
        ss += hv * a[f];
        sd += hv * a[F_OUT + f];
    }
    #pragma unroll
    for (int off = 16; off > 0; off >>= 1) {
        ss += __shfl_xor(ss, off, 32);
        sd += __shfl_xor(sd, off, 32);
    }
    if (lane == 0) { esrc[node] = ss; edst[node] = sd; }
}

// ---------------------------------------------------------------------------
// Kernel 3: zero the accumulators (out is poisoned and accumulated in place).
// ---------------------------------------------------------------------------
__global__ void gat_zero(float* __restrict__ out, float* __restrict__ rowsum,
                         int n_out, int n_rs) {
    const int i = blockIdx.x * blockDim.x + threadIdx.x;
    if (i < n_out) out[i] = 0.f;
    if (i < n_rs)  rowsum[i] = 0.f;
}

// ---------------------------------------------------------------------------
// Kernel 4: per-block max of leakyrelu(e_src[src]+e_dst[dst]) over edges.
// ---------------------------------------------------------------------------
__global__ __launch_bounds__(256) void gat_edge_max(const long long* __restrict__ edge,
                                                    const float* __restrict__ esrc,
                                                    const float* __restrict__ edst,
                                                    float* __restrict__ bmax, int E) {
    __shared__ float sm[256];
    const int tid = threadIdx.x;
    float m = -3.402823466e38f;
    for (int k = blockIdx.x * 256 + tid; k < E; k += gridDim.x * 256) {
        float e = esrc[(int)edge[k]] + edst[(int)edge[(size_t)E + k]];
        e = (e > 0.f) ? e : ALPHA * e;
        m = fmaxf(m, e);
    }
    sm[tid] = m;
    __syncthreads();
    for (int s = 128; s > 0; s >>= 1) {
        if (tid < s) sm[tid] = fmaxf(sm[tid], sm[tid + s]);
        __syncthreads();
    }
    if (tid == 0) bmax[blockIdx.x] = sm[0];
}

__global__ void gat_max_reduce(const float* __restrict__ bmax,
                               float* __restrict__ gmax, int nb) {
    __shared__ float sm[1024];
    const int tid = threadIdx.x;
    sm[tid] = (tid < nb) ? bmax[tid] : -3.402823466e38f;
    __syncthreads();
    for (int s = 512; s > 0; s >>= 1) {
        if (tid < s) sm[tid] = fmaxf(sm[tid], sm[tid + s]);
        __syncthreads();
    }
    if (tid == 0) *gmax = sm[0];
}

// ---------------------------------------------------------------------------
// Kernel 5: edge scatter.  One wave per edge; h[dst] gather + f32 HW atomics
// into out[src] (both buffers are L2-resident: 102 MB each vs 192 MB L2).
// ---------------------------------------------------------------------------
__global__ __launch_bounds__(256) void gat_scatter(const long long* __restrict__ edge,
                                                   const float* __restrict__ esrc,
                                                   const float* __restrict__ edst,
                                                   const float* __restrict__ gmax,
                                                   const float* __restrict__ h,
                                                   float* __restrict__ out,
                                                   float* __restrict__ rowsum, int E) {
    const int k = blockIdx.x * 8 + (threadIdx.x >> 5);
    if (k >= E) return;
    const int lane = threadIdx.x & 31;
    const int s = (int)edge[k];
    const int d = (int)edge[(size_t)E + k];
    float e = esrc[s] + edst[d];
    e = (e > 0.f) ? e : ALPHA * e;
    const float ev = __expf(e - *gmax);
    const float* hd = h + (size_t)d * F_OUT;
    float* os = out + (size_t)s * F_OUT;
    #pragma unroll
    for (int t = 0; t < 8; ++t) {
        const int f = lane + 32 * t;
        unsafeAtomicAdd(&os[f], ev * hd[f]);
    }
    if (lane == 0) unsafeAtomicAdd(&rowsum[s], ev);
}

// ---------------------------------------------------------------------------
// Kernel 6: normalize + ELU.
// ---------------------------------------------------------------------------
__global__ void gat_finalize(float* __restrict__ out,
                             const float* __restrict__ rowsum, int total) {
    const int i = blockIdx.x * blockDim.x + threadIdx.x;
    if (i >= total) return;
    const float v = out[i] / (rowsum[i >> 8] + GAT_EPS);
    out[i] = (v > 0.f) ? v : (__expf(v) - 1.f);
}

// ---------------------------------------------------------------------------
extern "C" void kernel_launch(void* const* d_in, const int* in_sizes, int n_in,
                              void* d_out, int out_size, void* d_ws, size_t ws_size,
                              hipStream_t stream) {
    const float*     x    = (const float*)d_in[0];
    const long long* edge = (const long long*)d_in[1];
    const float*     W    = (const float*)d_in[2];
    const float*     a    = (const float*)d_in[3];
    const float*     bias = (const float*)d_in[4];
    float* out = (float*)d_out;

    const int N = in_sizes[0] / F_IN;    // 100000 (multiple of 16)
    const int E = in_sizes[1] / 2;       // 3200000 (multiple of 8)

    // Workspace layout (~104 MB)
    float* h      = (float*)d_ws;                 // N*256
    float* esrc   = h + (size_t)N * F_OUT;        // N
    float* edst   = esrc + N;                     // N
    float* rowsum = edst + N;                     // N
    float* bmax   = rowsum + N;                   // 1024
    float* gmax   = bmax + 1024;                  // 1

    gat_gemm<<<N / 16, 256, 0, stream>>>(x, W, bias, h);
    gat_coef<<<(N + 7) / 8, 256, 0, stream>>>(h, a, esrc, edst, N);
    gat_zero<<<(N * F_OUT + 255) / 256, 256, 0, stream>>>(out, rowsum, N * F_OUT, N);
    gat_edge_max<<<1024, 256, 0, stream>>>(edge, esrc, edst, bmax, E);
    gat_max_reduce<<<1, 1024, 0, stream>>>(bmax, gmax, 1024);
    gat_scatter<<<(E + 7) / 8, 256, 0, stream>>>(edge, esrc, edst, gmax, h, out, rowsum, E);
    gat_finalize<<<(N * F_OUT + 255) / 256, 256, 0, stream>>>(out, rowsum, N * F_OUT);
}